// HoltWintersDecomposition_45019847196804
// MI455X (gfx1250) — compile-verified
//
#include <hip/hip_runtime.h>
#include <hip/hip_bf16.h>
#include <math.h>

#define B_  64
#define T_  2048
#define D_  32
#define E_  170
#define L_  24
#define BT_ (B_ * T_)          // 131072 rows
#define MTILES (BT_ / 16)      // 8192
#define NTILES 11              // ceil(170/16)

typedef __attribute__((ext_vector_type(2))) float v2f;
typedef __attribute__((ext_vector_type(8))) float v8f;

// ---------------------------------------------------------------------------
// Phase 1: Holt-Winters recurrence. One thread per (b,d) sequence, lane = d so
// x / level / trend / season accesses are 128B-coalesced per wave. The 24-slot
// season ring buffer lives in registers: the t-loop is unrolled by exactly 24
// so buf[(1+j)%24] is a compile-time index for every unrolled slot.
// x is read once -> non-temporal loads (don't waste L2 on it). The lev/tr/sea
// intermediates are re-read 11x by the GEMM -> regular temporal stores so the
// 48 MB stays resident in the 192 MB L2.
// ---------------------------------------------------------------------------
__global__ __launch_bounds__(64) void hw_scan_kernel(
    const float* __restrict__ x,
    const float* __restrict__ alpha_p,
    const float* __restrict__ beta_p,
    const float* __restrict__ gamma_p,
    float* __restrict__ lev_out,
    float* __restrict__ tr_out,
    float* __restrict__ sea_out)
{
    const int tid = blockIdx.x * blockDim.x + threadIdx.x;   // 0..2047
    const int b = tid >> 5;
    const int d = tid & 31;

    const float a  = 1.0f / (1.0f + expf(-alpha_p[0]));
    const float bb = 1.0f / (1.0f + expf(-beta_p[0]));
    const float g  = 1.0f / (1.0f + expf(-gamma_p[0]));
    const float ia = 1.0f - a, ib = 1.0f - bb, ig = 1.0f - g;

    const size_t base = (size_t)b * T_ * D_ + d;
    const float* __restrict__ xp = x       + base;
    float* __restrict__ lp       = lev_out + base;
    float* __restrict__ tp       = tr_out  + base;
    float* __restrict__ sp       = sea_out + base;

    float lev = __builtin_nontemporal_load(xp);
    float tr  = 0.0f;
    float buf[L_];
#pragma unroll
    for (int i = 0; i < L_; ++i) buf[i] = 0.0f;

    // t = 0 outputs: level0 = x[:,0], trend0 = 0, season0 = 0 (ref concat)
    lp[0] = lev;
    tp[0] = 0.0f;
    sp[0] = 0.0f;

    // Main loop: t = 1 .. 2040 in 85 blocks of 24 (pos pattern repeats).
    int t = 1;
    for (int blk = 0; blk < 85; ++blk) {
#pragma unroll
        for (int j = 0; j < L_; ++j) {
            const int pos = (1 + j) % L_;          // compile-time constant
            const float xt = __builtin_nontemporal_load(xp + (size_t)t * D_);
            const float ps = buf[pos];
            const float lt = a * (xt - ps) + ia * (lev + tr);
            const float tn = bb * (lt - lev) + ib * tr;
            const float se = g * (xt - lt) + ig * ps;
            buf[pos] = se;
            lev = lt; tr = tn;
            const size_t off = (size_t)t * D_;
            lp[off] = lt;
            tp[off] = tn;
            sp[off] = se;
            ++t;
        }
    }
    // Remainder: t = 2041..2047, pos = 1..7
#pragma unroll
    for (int j = 0; j < 7; ++j) {
        const int pos = 1 + j;
        const float xt = __builtin_nontemporal_load(xp + (size_t)t * D_);
        const float ps = buf[pos];
        const float lt = a * (xt - ps) + ia * (lev + tr);
        const float tn = bb * (lt - lev) + ib * tr;
        const float se = g * (xt - lt) + ig * ps;
        buf[pos] = se;
        lev = lt; tr = tn;
        const size_t off = (size_t)t * D_;
        lp[off] = lt;
        tp[off] = tn;
        sp[off] = se;
        ++t;
    }
}

// ---------------------------------------------------------------------------
// Phase 2: (BT x 32) x (32 x 170)^T via V_WMMA_F32_16X16X4_F32, K accumulated
// over 8 WMMA steps. One wave per 16x16 output tile. blockIdx.z selects
// level / trend / season. Intermediates (48 MB) are L2-resident; the 267 MB
// output is write-once -> non-temporal stores so it streams to HBM without
// evicting the intermediates from L2.
// ---------------------------------------------------------------------------
__global__ __launch_bounds__(256) void enc_gemm_kernel(
    const float* __restrict__ lev,
    const float* __restrict__ tr,
    const float* __restrict__ sea,
    const float* __restrict__ lw,
    const float* __restrict__ tw,
    const float* __restrict__ sw,
    const float* __restrict__ lb,
    const float* __restrict__ tb,
    const float* __restrict__ sb,
    float* __restrict__ out)
{
    const int wave = threadIdx.x >> 5;
    const int lane = threadIdx.x & 31;
    const int mtile = blockIdx.x * 8 + wave;       // 0..8191
    const int n0 = blockIdx.y * 16;                // 0,16,...,160
    const int z = blockIdx.z;                      // 0=level 1=trend 2=season

    const float* A    = (z == 0) ? lev : (z == 1) ? tr : sea;
    const float* W    = (z == 0) ? lw  : (z == 1) ? tw : sw;
    const float* bias = (z == 0) ? lb  : (z == 1) ? tb : sb;
    float* O = out + (size_t)z * BT_ * E_;

    const int col = lane & 15;
    const int hi  = lane >> 4;

    // A fragment: lane holds row (m0 + col), K pair at (k + hi*2)
    const float* arow = A + (size_t)(mtile * 16 + col) * D_ + hi * 2;
    // B fragment: B[k'][n] = W[n][k']; lane holds weight row (n0 + col)
    const int wr = (n0 + col < E_) ? (n0 + col) : (E_ - 1);
    const float* wrow = W + (size_t)wr * D_ + hi * 2;

    v8f acc = {};
#pragma unroll
    for (int k = 0; k < D_; k += 4) {
        v2f av = *(const v2f*)(arow + k);
        v2f bv = *(const v2f*)(wrow + k);
        // D = A(16x4 f32) * B(4x16 f32) + C, full fp32
        acc = __builtin_amdgcn_wmma_f32_16x16x4_f32(
            /*neg_a=*/false, av, /*neg_b=*/false, bv,
            /*c_mod=*/(short)0, acc, /*reuse_a=*/false, /*reuse_b=*/false);
    }

    // D layout: VGPR r -> row (r + 8*hi), col = lane&15
    const int n = n0 + col;
    if (n < E_) {
        const float bv = bias[n];
        float* op = O + (size_t)(mtile * 16 + hi * 8) * E_ + n;
#pragma unroll
        for (int r = 0; r < 8; ++r)
            __builtin_nontemporal_store(acc[r] + bv, op + (size_t)r * E_);
    }
}

extern "C" void kernel_launch(void* const* d_in, const int* in_sizes, int n_in,
                              void* d_out, int out_size, void* d_ws, size_t ws_size,
                              hipStream_t stream) {
    const float* x        = (const float*)d_in[0];
    const float* alpha    = (const float*)d_in[1];
    const float* beta     = (const float*)d_in[2];
    const float* gamma    = (const float*)d_in[3];
    const float* level_w  = (const float*)d_in[4];
    const float* level_b  = (const float*)d_in[5];
    const float* trend_w  = (const float*)d_in[6];
    const float* trend_b  = (const float*)d_in[7];
    const float* season_w = (const float*)d_in[8];
    const float* season_b = (const float*)d_in[9];

    float* ws  = (float*)d_ws;
    const size_t plane = (size_t)B_ * T_ * D_;     // 4,194,304 floats (16 MB)
    float* lev = ws;
    float* tr  = ws + plane;
    float* sea = ws + 2 * plane;

    hw_scan_kernel<<<dim3(32), dim3(64), 0, stream>>>(
        x, alpha, beta, gamma, lev, tr, sea);

    dim3 grid(MTILES / 8, NTILES, 3);
    dim3 block(256);
    enc_gemm_kernel<<<grid, block, 0, stream>>>(
        lev, tr, sea, level_w, trend_w, season_w,
        level_b, trend_b, season_b, (float*)d_out);
}